// NumericEmbedding_54331336294524
// MI455X (gfx1250) — compile-verified
//
#include <hip/hip_runtime.h>

typedef __attribute__((ext_vector_type(2))) float v2f;
typedef __attribute__((ext_vector_type(4))) float v4f;
typedef __attribute__((ext_vector_type(8))) float v8f;

#define TARGET_DIM 768
#define FLUX_DIM   256
#define N_POS      (32 * 4096)     // BATCH * SEQ = 131072
#define ROWS_PER_BLOCK 16

// ---------------------------------------------------------------------------
// Kernel 1: fused GEMV via fp32 WMMA.
//   alpha[d] = sum_f W_proj[d,f] * w_enc[f]
//   beta[d]  = sum_f W_proj[d,f] * b_enc[f] + b_proj[d]
// One wave per 16-row chunk of d; K advanced 4/step with V_WMMA_F32_16X16X4_F32.
// B has w_enc in column 0, b_enc in column 1, zeros elsewhere (built with
// branch-free arithmetic masks so EXEC stays all-1s and no divergent loads),
// so D[:,0] = alpha tile and D[:,1] = beta-partial tile.
// ---------------------------------------------------------------------------
__global__ void __launch_bounds__(32)
numemb_precompute_wmma(const float* __restrict__ W_proj,  // [768,256] row-major
                       const float* __restrict__ w_enc,   // [256]
                       const float* __restrict__ b_enc,   // [256]
                       const float* __restrict__ b_proj,  // [768]
                       float* __restrict__ alpha,         // [768]
                       float* __restrict__ beta)          // [768]
{
    const int lane  = threadIdx.x;        // 0..31, all lanes active (EXEC=~0)
    const int d0    = blockIdx.x * 16;    // 48 blocks cover 768 rows
    const int row   = lane & 15;          // A-matrix M index / B-matrix N index
    const int khalf = (lane >> 4) << 1;   // 0 for lanes 0-15, 2 for lanes 16-31
    const int col   = row;                // B/D column this lane owns

    // Branch-free column selectors (computed once; keeps the K-loop straight-line)
    const float mw = (col == 0) ? 1.0f : 0.0f;   // column 0 <- w_enc
    const float me = (col == 1) ? 1.0f : 0.0f;   // column 1 <- b_enc

    v8f acc = {0.f, 0.f, 0.f, 0.f, 0.f, 0.f, 0.f, 0.f};
    const float* __restrict__ wrow = W_proj + (size_t)(d0 + row) * FLUX_DIM;

    for (int k0 = 0; k0 < FLUX_DIM; k0 += 4) {
        // A tile (16x4 f32): VGPR0 = K=khalf, VGPR1 = K=khalf+1
        v2f a = *(const v2f*)(wrow + k0 + khalf);
        // Unconditional pair loads of the encoder vectors at this K slice
        v2f w = *(const v2f*)(w_enc + k0 + khalf);
        v2f e = *(const v2f*)(b_enc + k0 + khalf);
        // B tile (4x16 f32): lane owns column `col`; mask-select, no branches
        v2f b;
        b.x = w.x * mw + e.x * me;
        b.y = w.y * mw + e.y * me;
        // D = A*B + C  (8 args: neg_a, A, neg_b, B, c_mod, C, reuse_a, reuse_b)
        acc = __builtin_amdgcn_wmma_f32_16x16x4_f32(
            false, a, false, b, (short)0, acc, false, false);
    }

    // D layout: VGPR v, lanes 0-15 -> (M=v, N=lane); lanes 16-31 -> (M=v+8, N=lane-16)
    const int mbase = d0 + ((lane >> 4) << 3);
    if (col == 0) {
#pragma unroll
        for (int v = 0; v < 8; ++v) alpha[mbase + v] = acc[v];
    } else if (col == 1) {
#pragma unroll
        for (int v = 0; v < 8; ++v) beta[mbase + v] = acc[v] + b_proj[mbase + v];
    }
}

// ---------------------------------------------------------------------------
// Kernel 2: store-bandwidth-bound rank-1 broadcast.
//   out[p, d] = x[p] * alpha[d] + beta[d]
// 256 threads/block (8 wave32). Thread t owns float4 columns c, c+64, c+128
// (c = t&63) and row-subgroup t>>6; alpha/beta float4s live in registers for
// all 16 rows of the block. Non-temporal b128 stores (403 MB > 192 MB L2).
// ---------------------------------------------------------------------------
__global__ void __launch_bounds__(256)
numemb_broadcast_fma(const float* __restrict__ x,      // [131072]
                     const float* __restrict__ alpha,  // [768]
                     const float* __restrict__ beta,   // [768]
                     float* __restrict__ out)          // [131072, 768]
{
    const int t    = threadIdx.x;
    const int c    = t & 63;    // float4 column index within 0..191 (3 groups of 64)
    const int rsub = t >> 6;    // 0..3
    const size_t p0 = (size_t)blockIdx.x * ROWS_PER_BLOCK;

    const v4f* __restrict__ a4 = (const v4f*)alpha;
    const v4f* __restrict__ b4 = (const v4f*)beta;
    const v4f A0 = a4[c], A1 = a4[c + 64], A2 = a4[c + 128];
    const v4f B0 = b4[c], B1 = b4[c + 64], B2 = b4[c + 128];

#pragma unroll
    for (int r = 0; r < ROWS_PER_BLOCK / 4; ++r) {
        const size_t p  = p0 + rsub + 4 * r;
        const float xv  = x[p];
        v4f o0 = A0 * xv + B0;  // v_fma per component
        v4f o1 = A1 * xv + B1;
        v4f o2 = A2 * xv + B2;
        v4f* __restrict__ orow = (v4f*)(out + p * TARGET_DIM);
        __builtin_nontemporal_store(o0, orow + c);
        __builtin_nontemporal_store(o1, orow + c + 64);
        __builtin_nontemporal_store(o2, orow + c + 128);
    }
}

// ---------------------------------------------------------------------------
extern "C" void kernel_launch(void* const* d_in, const int* in_sizes, int n_in,
                              void* d_out, int out_size, void* d_ws, size_t ws_size,
                              hipStream_t stream) {
    const float* numbers = (const float*)d_in[0];  // [32, 4096]
    const float* w_enc   = (const float*)d_in[1];  // [256]
    const float* b_enc   = (const float*)d_in[2];  // [256]
    const float* W_proj  = (const float*)d_in[3];  // [768, 256]
    const float* b_proj  = (const float*)d_in[4];  // [768]
    float* out   = (float*)d_out;                  // [32, 4096, 768]
    float* alpha = (float*)d_ws;                   // ws[0 .. 768)
    float* beta  = alpha + TARGET_DIM;             // ws[768 .. 1536)

    // 1) alpha/beta via fp32 WMMA: 48 chunks of 16 rows, one wave each.
    numemb_precompute_wmma<<<TARGET_DIM / 16, 32, 0, stream>>>(
        W_proj, w_enc, b_enc, b_proj, alpha, beta);

    // 2) rank-1 broadcast: 131072 rows / 16 per block = 8192 blocks.
    numemb_broadcast_fma<<<N_POS / ROWS_PER_BLOCK, 256, 0, stream>>>(
        numbers, alpha, beta, out);
}